// NeuralODE_77232101917084
// MI455X (gfx1250) — compile-verified
//
#include <hip/hip_runtime.h>
#include <math.h>

typedef float v2f __attribute__((ext_vector_type(2)));
typedef float v8f __attribute__((ext_vector_type(8)));

namespace {
constexpr int Bn = 4096, Tn = 101, Dn = 64, Hn = 64;
constexpr int ROWS = 32;          // batch rows per workgroup (2 N-tiles)
constexpr int PS   = 68;          // floats per feature-PAIR row: 2*ROWS + 4 pad
constexpr int KS64 = 66;          // padded row stride for K=64 weights
constexpr int KS32 = 34;          // padded row stride for K=32 weights
constexpr int NTHREADS = 128;     // 4 waves (wave32)

// Tsit5 tableau
constexpr float cA21 = 0.161f;
constexpr float cA31 = -0.008480655492356989f, cA32 = 0.335480655492357f;
constexpr float cA41 = 2.8971530571054935f, cA42 = -6.359448489975075f, cA43 = 4.3622954328695815f;
constexpr float cA51 = 5.325864828439257f, cA52 = -11.748883564062828f, cA53 = 7.4955393428898365f, cA54 = -0.09249506636175525f;
constexpr float cA61 = 5.86145544294642f, cA62 = -12.92096931784711f, cA63 = 8.159367898576159f, cA64 = -0.071584973281401f, cA65 = -0.028269050394068383f;
constexpr float cB1 = 0.09646076681806523f, cB2 = 0.01f, cB3 = 0.4798896504144996f;
constexpr float cB4 = 1.379008574103742f, cB5 = -3.290069515436081f, cB6 = 2.324710524099774f;
}

__device__ __forceinline__ float act_tanh(float x) {
#if __has_builtin(__builtin_amdgcn_tanhf)
  return __builtin_amdgcn_tanhf(x);      // v_tanh_f32 (CDNA5 TRANS op)
#else
  return tanhf(x);
#endif
}

// Full-precision fp32 WMMA: D = A(16x4) * B(4x16) + C(16x16)
__device__ __forceinline__ v8f wmma4(v2f a, v2f b, v8f c) {
  return __builtin_amdgcn_wmma_f32_16x16x4_f32(
      /*neg_a=*/false, a, /*neg_b=*/false, b,
      /*c_mod=*/(short)0, c, /*reuse_a=*/false, /*reuse_b=*/false);
}

// Activation layout: feature-pair interleaved.
//   element (feat f, batch n) lives at  (f>>1)*PS + 2*n + (f&1)
// -> WMMA B operand for K-slice k0=4s, lane half h, col n is ONE aligned
//    float2 at pair-row (2s+h), offset 2n  ({k even, k odd} = {VGPR0, VGPR1}).
// -> WMMA C writeback: VGPRs (2r',2r'+1) hold consecutive rows -> float2 store
//    at pair-row m0/2 + r' + 4h, offset 2n.

__global__ __launch_bounds__(NTHREADS) void node_tsit5(
    const float* __restrict__ ts, const float* __restrict__ y0,
    const float* __restrict__ encW, const float* __restrict__ encb,
    const float* __restrict__ gW1, const float* __restrict__ gb1,
    const float* __restrict__ gW2, const float* __restrict__ gb2,
    const float* __restrict__ gW3, const float* __restrict__ gb3,
    const float* __restrict__ decW, const float* __restrict__ decb,
    float* __restrict__ ys, float* __restrict__ zs)
{
  // ---- LDS ----
  __shared__ float sWe[Hn * KS64];        // enc_W [64][64], padded rows
  __shared__ float sWd[Dn * KS64];        // dec_W [64][64]
  __shared__ float sW1[32 * KS64];        // W1 [32][64]
  __shared__ float sW2[32 * KS32];        // W2 [32][32]
  __shared__ float sW3[64 * KS32];        // W3 [64][32]
  __shared__ float sbe[64], sbd[64], sb1[32], sb2[32], sb3[64];
  __shared__ float sZ[32 * PS];           // state z   (64 feats = 32 pair-rows)
  __shared__ float sY[32 * PS];           // Tsit5 stage input
  __shared__ float sHa[16 * PS];          // MLP hidden 1 (32 feats)
  __shared__ float sHb[16 * PS];          // MLP hidden 2
  __shared__ float sK[6][32 * PS];        // k1..k6

  const int tid  = threadIdx.x;
  const int lane = tid & 31;
  const int wave = tid >> 5;
  const int ln   = lane & 15;             // N (or M) index within a tile
  const int half = lane >> 4;             // lane half selects K/M offsets
  const int bbase = blockIdx.x * ROWS;

  // ---- stage weights into LDS ----
  for (int i = tid; i < 64 * 64; i += NTHREADS) {
    const int r = i >> 6, c = i & 63;
    sWe[r * KS64 + c] = encW[i];
    sWd[r * KS64 + c] = decW[i];
  }
  for (int i = tid; i < 32 * 64; i += NTHREADS) { const int r = i >> 6, c = i & 63; sW1[r * KS64 + c] = gW1[i]; }
  for (int i = tid; i < 32 * 32; i += NTHREADS) { const int r = i >> 5, c = i & 31; sW2[r * KS32 + c] = gW2[i]; }
  for (int i = tid; i < 64 * 32; i += NTHREADS) { const int r = i >> 5, c = i & 31; sW3[r * KS32 + c] = gW3[i]; }
  if (tid < 64) { sbe[tid] = encb[tid]; sbd[tid] = decb[tid]; sb3[tid] = gb3[tid]; }
  if (tid < 32) { sb1[tid] = gb1[tid]; sb2[tid] = gb2[tid]; }
  // y0 tile -> pair-interleaved feature-major (coalesced global reads)
  for (int i = tid; i < ROWS * 64; i += NTHREADS) {
    const int bl = i >> 6, d = i & 63;
    sY[(d >> 1) * PS + 2 * bl + (d & 1)] = y0[(size_t)(bbase + bl) * Dn + d];
  }
  __syncthreads();

  const float dt = (ts[Tn - 1] - ts[0]) / (float)(Tn - 1);

  // job assignment (uniform per wave -> EXEC all-ones around WMMA)
  const int m01 = (wave >> 1) * 16;       // L1/L2 M-tile
  const int n01 = (wave & 1) * 16;        // L1/L2 N-tile
  const int m03 = wave * 16;              // L3 / enc / dec M-tile

  // ---- hoist loop-invariant weights into registers (A-operand layout) ----
  v2f aW1r[16], aW2r[8], aW3r[8], aWdr[16];
#pragma unroll
  for (int s = 0; s < 16; ++s) aW1r[s] = *(const v2f*)&sW1[(m01 + ln) * KS64 + 4 * s + 2 * half];
#pragma unroll
  for (int s = 0; s < 8; ++s)  aW2r[s] = *(const v2f*)&sW2[(m01 + ln) * KS32 + 4 * s + 2 * half];
#pragma unroll
  for (int s = 0; s < 8; ++s)  aW3r[s] = *(const v2f*)&sW3[(m03 + ln) * KS32 + 4 * s + 2 * half];
#pragma unroll
  for (int s = 0; s < 16; ++s) aWdr[s] = *(const v2f*)&sWd[(m03 + ln) * KS64 + 4 * s + 2 * half];

  // ---- encode: z0^T = enc_W @ y0^T + enc_b  (input sY, output sZ) ----
#pragma unroll
  for (int nt = 0; nt < 2; ++nt) {
    const int n0 = nt * 16;
    v8f c = {};
#pragma unroll
    for (int s = 0; s < 16; ++s) {
      const v2f a = *(const v2f*)&sWe[(m03 + ln) * KS64 + 4 * s + 2 * half];
      const v2f b = *(const v2f*)&sY[(2 * s + half) * PS + 2 * (n0 + ln)];
      c = wmma4(a, b, c);
    }
#pragma unroll
    for (int r = 0; r < 4; ++r) {
      const int m = m03 + 2 * r + 8 * half;
      const v2f bias = *(const v2f*)&sbe[m];
      v2f v; v.x = c[2 * r] + bias.x; v.y = c[2 * r + 1] + bias.y;
      *(v2f*)&sZ[(m >> 1) * PS + 2 * (n0 + ln)] = v;
    }
  }
  __syncthreads();

  // ---- decode + store zs/ys for time index t (reads sZ only) ----
  auto emit = [&](int t) {
#pragma unroll
    for (int nt = 0; nt < 2; ++nt) {
      const int n0 = nt * 16;
      v8f c = {};
#pragma unroll
      for (int s = 0; s < 16; ++s) {
        const v2f b = *(const v2f*)&sZ[(2 * s + half) * PS + 2 * (n0 + ln)];
        c = wmma4(aWdr[s], b, c);
      }
      const size_t bg = (size_t)(bbase + n0 + ln);
#pragma unroll
      for (int r = 0; r < 4; ++r) {
        const int m = m03 + 2 * r + 8 * half;
        const v2f bias = *(const v2f*)&sbd[m];
        v2f v; v.x = c[2 * r] + bias.x; v.y = c[2 * r + 1] + bias.y;
        *(v2f*)&ys[(bg * Tn + t) * Dn + m] = v;
      }
    }
    for (int i = tid; i < ROWS * 32; i += NTHREADS) {     // zs, float2 per job
      const int p = i & 31, bl = i >> 5;                  // p = feature pair
      const v2f v = *(const v2f*)&sZ[p * PS + 2 * bl];
      *(v2f*)&zs[((size_t)(bbase + bl) * Tn + t) * Hn + 2 * p] = v;
    }
  };

  // ---- MLP vector field ----
  auto mlp = [&](const float* in, float* out) {
    { // layer 1: K=64 -> sHa  (one 16x16 job per wave)
      v8f c = {};
#pragma unroll
      for (int s = 0; s < 16; ++s) {
        const v2f b = *(const v2f*)&in[(2 * s + half) * PS + 2 * (n01 + ln)];
        c = wmma4(aW1r[s], b, c);
      }
#pragma unroll
      for (int r = 0; r < 4; ++r) {
        const int m = m01 + 2 * r + 8 * half;
        const v2f bias = *(const v2f*)&sb1[m];
        v2f v; v.x = act_tanh(c[2 * r] + bias.x); v.y = act_tanh(c[2 * r + 1] + bias.y);
        *(v2f*)&sHa[(m >> 1) * PS + 2 * (n01 + ln)] = v;
      }
    }
    __syncthreads();
    { // layer 2: K=32 -> sHb
      v8f c = {};
#pragma unroll
      for (int s = 0; s < 8; ++s) {
        const v2f b = *(const v2f*)&sHa[(2 * s + half) * PS + 2 * (n01 + ln)];
        c = wmma4(aW2r[s], b, c);
      }
#pragma unroll
      for (int r = 0; r < 4; ++r) {
        const int m = m01 + 2 * r + 8 * half;
        const v2f bias = *(const v2f*)&sb2[m];
        v2f v; v.x = act_tanh(c[2 * r] + bias.x); v.y = act_tanh(c[2 * r + 1] + bias.y);
        *(v2f*)&sHb[(m >> 1) * PS + 2 * (n01 + ln)] = v;
      }
    }
    __syncthreads();
    // layer 3: K=32, 64 out-features (two 16x16 jobs per wave)
#pragma unroll
    for (int nt = 0; nt < 2; ++nt) {
      const int n0 = nt * 16;
      v8f c = {};
#pragma unroll
      for (int s = 0; s < 8; ++s) {
        const v2f b = *(const v2f*)&sHb[(2 * s + half) * PS + 2 * (n0 + ln)];
        c = wmma4(aW3r[s], b, c);
      }
#pragma unroll
      for (int r = 0; r < 4; ++r) {
        const int m = m03 + 2 * r + 8 * half;
        const v2f bias = *(const v2f*)&sb3[m];
        v2f v; v.x = c[2 * r] + bias.x; v.y = c[2 * r + 1] + bias.y;
        *(v2f*)&out[(m >> 1) * PS + 2 * (n0 + ln)] = v;
      }
    }
    __syncthreads();
  };

  float* k1 = sK[0]; float* k2 = sK[1]; float* k3 = sK[2];
  float* k4 = sK[3]; float* k5 = sK[4]; float* k6 = sK[5];

  // elementwise Tsit5 combines on float2 feature-pairs (1024 jobs)
  auto pr = [&](const float* b, int i) -> v2f { return *(const v2f*)&b[(i >> 5) * PS + 2 * (i & 31)]; };
  auto pw = [&](float* b, int i, v2f v) { *(v2f*)&b[(i >> 5) * PS + 2 * (i & 31)] = v; };

  emit(0);

  for (int t = 1; t < Tn; ++t) {
    mlp(sZ, k1);
    for (int i = tid; i < 32 * ROWS; i += NTHREADS) {
      const v2f z = pr(sZ, i), v1 = pr(k1, i);
      v2f y; y.x = fmaf(dt, cA21 * v1.x, z.x); y.y = fmaf(dt, cA21 * v1.y, z.y);
      pw(sY, i, y);
    }
    __syncthreads();
    mlp(sY, k2);
    for (int i = tid; i < 32 * ROWS; i += NTHREADS) {
      const v2f z = pr(sZ, i), v1 = pr(k1, i), v2 = pr(k2, i);
      v2f y;
      y.x = fmaf(dt, fmaf(cA31, v1.x, cA32 * v2.x), z.x);
      y.y = fmaf(dt, fmaf(cA31, v1.y, cA32 * v2.y), z.y);
      pw(sY, i, y);
    }
    __syncthreads();
    mlp(sY, k3);
    for (int i = tid; i < 32 * ROWS; i += NTHREADS) {
      const v2f z = pr(sZ, i), v1 = pr(k1, i), v2 = pr(k2, i), v3 = pr(k3, i);
      v2f y;
      y.x = fmaf(dt, fmaf(cA41, v1.x, fmaf(cA42, v2.x, cA43 * v3.x)), z.x);
      y.y = fmaf(dt, fmaf(cA41, v1.y, fmaf(cA42, v2.y, cA43 * v3.y)), z.y);
      pw(sY, i, y);
    }
    __syncthreads();
    mlp(sY, k4);
    for (int i = tid; i < 32 * ROWS; i += NTHREADS) {
      const v2f z = pr(sZ, i), v1 = pr(k1, i), v2 = pr(k2, i), v3 = pr(k3, i), v4 = pr(k4, i);
      v2f y;
      y.x = fmaf(dt, fmaf(cA51, v1.x, fmaf(cA52, v2.x, fmaf(cA53, v3.x, cA54 * v4.x))), z.x);
      y.y = fmaf(dt, fmaf(cA51, v1.y, fmaf(cA52, v2.y, fmaf(cA53, v3.y, cA54 * v4.y))), z.y);
      pw(sY, i, y);
    }
    __syncthreads();
    mlp(sY, k5);
    for (int i = tid; i < 32 * ROWS; i += NTHREADS) {
      const v2f z = pr(sZ, i), v1 = pr(k1, i), v2 = pr(k2, i), v3 = pr(k3, i), v4 = pr(k4, i), v5 = pr(k5, i);
      v2f y;
      y.x = fmaf(dt, fmaf(cA61, v1.x, fmaf(cA62, v2.x, fmaf(cA63, v3.x, fmaf(cA64, v4.x, cA65 * v5.x)))), z.x);
      y.y = fmaf(dt, fmaf(cA61, v1.y, fmaf(cA62, v2.y, fmaf(cA63, v3.y, fmaf(cA64, v4.y, cA65 * v5.y)))), z.y);
      pw(sY, i, y);
    }
    __syncthreads();
    mlp(sY, k6);
    for (int i = tid; i < 32 * ROWS; i += NTHREADS) {   // 5th-order solution
      const v2f z = pr(sZ, i), v1 = pr(k1, i), v2 = pr(k2, i), v3 = pr(k3, i),
                v4 = pr(k4, i), v5 = pr(k5, i), v6 = pr(k6, i);
      v2f zn;
      zn.x = fmaf(dt, fmaf(cB1, v1.x, fmaf(cB2, v2.x, fmaf(cB3, v3.x,
                   fmaf(cB4, v4.x, fmaf(cB5, v5.x, cB6 * v6.x))))), z.x);
      zn.y = fmaf(dt, fmaf(cB1, v1.y, fmaf(cB2, v2.y, fmaf(cB3, v3.y,
                   fmaf(cB4, v4.y, fmaf(cB5, v5.y, cB6 * v6.y))))), z.y);
      pw(sZ, i, zn);
    }
    __syncthreads();
    emit(t);
  }
}

extern "C" void kernel_launch(void* const* d_in, const int* in_sizes, int n_in,
                              void* d_out, int out_size, void* d_ws, size_t ws_size,
                              hipStream_t stream) {
  (void)in_sizes; (void)n_in; (void)out_size; (void)d_ws; (void)ws_size;
  const float* ts   = (const float*)d_in[0];
  const float* y0   = (const float*)d_in[1];
  const float* encW = (const float*)d_in[2];
  const float* encb = (const float*)d_in[3];
  const float* W1   = (const float*)d_in[4];
  const float* b1   = (const float*)d_in[5];
  const float* W2   = (const float*)d_in[6];
  const float* b2   = (const float*)d_in[7];
  const float* W3   = (const float*)d_in[8];
  const float* b3   = (const float*)d_in[9];
  const float* decW = (const float*)d_in[10];
  const float* decb = (const float*)d_in[11];
  float* ys = (float*)d_out;
  float* zs = ys + (size_t)Bn * Tn * Dn;
  dim3 grid(Bn / ROWS), block(NTHREADS);
  node_tsit5<<<grid, block, 0, stream>>>(ts, y0, encW, encb, W1, b1, W2, b2,
                                         W3, b3, decW, decb, ys, zs);
}